// UncompressTransformLayer_20100446945611
// MI455X (gfx1250) — compile-verified
//
#include <hip/hip_runtime.h>
#include <math.h>

// Symmetrize-from-packed: out (n x n, f32) = scatter(packed strict-upper) +
// transpose + identity.  Pure data movement: min traffic = 4*m read once
// (134 MB) + 4*n^2 written once (256 MB) ~= 390 MB -> ~17 us @ 23.3 TB/s.
//
// Strategy (MI455X):
//  * tile-pair blocks (bi<=bj): each packed element is read exactly ONCE,
//    coalesced (packed rows are contiguous); block writes both the tile and
//    its mirror -> avoids the stride-n gather a naive lower-triangle read
//    would do.
//  * CDNA5 async global->LDS staging (GLOBAL_LOAD_ASYNC_TO_LDS_B32 /
//    ASYNCcnt / s_wait_asynccnt) -- no VGPR round trip.
//  * LDS transpose with stride-33 padding (row-wise AND column-wise access
//    patterns are bank-conflict-free across a wave32).
//  * 128-bit nontemporal stores: stream the 256 MB output without polluting
//    L2 (output is written once, never re-read).

#define TILE 32

typedef __attribute__((ext_vector_type(4))) float v4f;

__global__ __launch_bounds__(256) void sym_uncompress_kernel(
    const float* __restrict__ comp, float* __restrict__ out, int n)
{
    __shared__ float tile[TILE][TILE + 1];

    const int bj = blockIdx.x;          // tile column
    const int bi = blockIdx.y;          // tile row
    if (bi > bj) return;                // only upper-triangular tile pairs work
    const bool diag = (bi == bj);

    const int tid  = threadIdx.x;       // 0..255 (8 waves of 32)
    const int lrow = tid >> 5;          // 0..7
    const int lcol = tid & 31;          // 0..31
    const int r0 = bi * TILE;
    const int c0 = bj * TILE;

    if (diag) {
        // Zero the LDS tile so upper + upper^T + I works uniformly.
        #pragma unroll
        for (int k = 0; k < 4; ++k)
            tile[lrow + 8 * k][lcol] = 0.0f;
        __syncthreads();   // zeros committed before async engine writes LDS
    }

    // --- async stage: packed global -> LDS (each element read once) ---
    #pragma unroll
    for (int k = 0; k < 4; ++k) {
        const int row = lrow + 8 * k;   // 0..31 within tile
        const int col = lcol;           // 0..31 within tile
        if (!diag || col > row) {
            const long long r = (long long)(r0 + row);
            // triu_indices(n, k=1) row-major index of (r, c):
            const long long idx =
                r * (long long)n - (r * (r + 1)) / 2 + (long long)(c0 + col) - r - 1;
            unsigned long long gaddr = (unsigned long long)(comp + idx);
            // flat addr of an LDS object: low 32 bits are the LDS byte address
            unsigned lds_off = (unsigned)(size_t)&tile[row][col];
            asm volatile("global_load_async_to_lds_b32 %0, %1, off"
                         :: "v"(lds_off), "v"(gaddr)
                         : "memory");
        }
    }
    asm volatile("s_wait_asynccnt 0" ::: "memory");  // our LDS writes landed
    __syncthreads();                                 // everyone's landed

    // --- stores: thread -> (row 0..31, one 4-wide column chunk) ---
    const int srow = tid >> 3;          // 0..31
    const int sc4  = (tid & 7) * 4;     // 0,4,...,28

    if (diag) {
        v4f v;
        #pragma unroll
        for (int k = 0; k < 4; ++k) {
            const int c = sc4 + k;
            v[k] = tile[srow][c] + tile[c][srow]
                 + (srow == c ? 1.0f : 0.0f);
        }
        v4f* p = (v4f*)(out + (size_t)(r0 + srow) * n + (c0 + sc4));
        __builtin_nontemporal_store(v, p);
    } else {
        v4f v;
        #pragma unroll
        for (int k = 0; k < 4; ++k)
            v[k] = tile[srow][sc4 + k];
        v4f* p = (v4f*)(out + (size_t)(r0 + srow) * n + (c0 + sc4));
        __builtin_nontemporal_store(v, p);

        v4f w;   // mirrored tile: transpose via LDS (conflict-free reads)
        #pragma unroll
        for (int k = 0; k < 4; ++k)
            w[k] = tile[sc4 + k][srow];
        v4f* q = (v4f*)(out + (size_t)(c0 + srow) * n + (r0 + sc4));
        __builtin_nontemporal_store(w, q);
    }
}

extern "C" void kernel_launch(void* const* d_in, const int* in_sizes, int n_in,
                              void* d_out, int out_size, void* d_ws, size_t ws_size,
                              hipStream_t stream) {
    (void)n_in; (void)d_ws; (void)ws_size; (void)out_size;
    const float* comp = (const float*)d_in[0];
    float* out = (float*)d_out;

    const long long m = (long long)in_sizes[0];                 // n*(n-1)/2
    const int n = (int)llround(sqrt(2.0 * (double)m)) + 1;      // 8192
    const int T = n / TILE;                                     // 256

    dim3 grid(T, T), block(256);
    sym_uncompress_kernel<<<grid, block, 0, stream>>>(comp, out, n);
}